// SMPL_55576876810724
// MI455X (gfx1250) — compile-verified
//
#include <hip/hip_runtime.h>

// ---- problem sizes --------------------------------------------------------
#define Bx   1024
#define Vx   6890
#define Jx   24
#define NBx  10
#define Px   207
#define FW   220      // padded feature width: 10 betas + 2 pad + 207 pose + 1 pad
#define NJOUT 61      // 24 chain + 11 extra + 9 extra9 + 17 h36m

__device__ __constant__ int d_parents[Jx] =
    {-1,0,0,0,1,2,3,4,5,6,7,8,9,9,9,12,13,14,16,17,18,19,20,21};

typedef float v2f __attribute__((ext_vector_type(2)));
typedef float v8f __attribute__((ext_vector_type(8)));

#define HAVE_WMMA_F32X4 __has_builtin(__builtin_amdgcn_wmma_f32_16x16x4_f32)

// ---------------------------------------------------------------------------
// K0: fold J_regressor into tiny coefficients:
//   Jt[j][k]    = sum_v Jr[j,v] * v_template[v,k]                (72 vals)
//   JS[j][k][l] = sum_v Jr[j,v] * shapedirs[v,k,l]               (720 vals)
// ---------------------------------------------------------------------------
__global__ void k_jcoef(const float* __restrict__ Jr,
                        const float* __restrict__ vt,
                        const float* __restrict__ sd,
                        float* __restrict__ Jt, float* __restrict__ JS) {
  __shared__ float red[256];
  const int o = blockIdx.x;              // 0..791
  float acc = 0.f;
  if (o < 72) {
    const int j = o / 3, k = o % 3;
    const float* row = Jr + (size_t)j * Vx;
    for (int v = threadIdx.x; v < Vx; v += 256)
      acc += row[v] * vt[v * 3 + k];
  } else {
    const int o2 = o - 72;
    const int j = o2 / 30, k = (o2 % 30) / 10, l = o2 % 10;
    const float* row = Jr + (size_t)j * Vx;
    for (int v = threadIdx.x; v < Vx; v += 256)
      acc += row[v] * sd[v * 30 + k * 10 + l];
  }
  red[threadIdx.x] = acc;
  __syncthreads();
  for (int s = 128; s > 0; s >>= 1) {
    if (threadIdx.x < s) red[threadIdx.x] += red[threadIdx.x + s];
    __syncthreads();
  }
  if (threadIdx.x == 0) {
    if (o < 72) Jt[o] = red[0]; else JS[o - 72] = red[0];
  }
}

// ---------------------------------------------------------------------------
// K1a: padded feature buffer F[b][220]: betas in [0..9], zeros elsewhere
// (pose entries [12..218] are overwritten by k_rodrigues afterwards)
// ---------------------------------------------------------------------------
__global__ void k_feat(const float* __restrict__ betas, float* __restrict__ F) {
  const int t = blockIdx.x * blockDim.x + threadIdx.x;
  if (t >= Bx * FW) return;
  const int b = t / FW, i = t % FW;
  F[t] = (i < NBx) ? betas[b * NBx + i] : 0.f;
}

// ---------------------------------------------------------------------------
// K1b: Rodrigues -> rot_mats (B,J,3,3); pose_feature into F[b][12 + q]
// ---------------------------------------------------------------------------
__global__ void k_rodrigues(const float* __restrict__ poses,
                            float* __restrict__ rot, float* __restrict__ F) {
  const int t = blockIdx.x * blockDim.x + threadIdx.x;
  if (t >= Bx * Jx) return;
  const int b = t / Jx, j = t % Jx;
  const float r0 = poses[b * 72 + j * 3 + 0];
  const float r1 = poses[b * 72 + j * 3 + 1];
  const float r2 = poses[b * 72 + j * 3 + 2];
  const float e0 = r0 + 1e-8f, e1 = r1 + 1e-8f, e2 = r2 + 1e-8f;
  const float angle = sqrtf(e0 * e0 + e1 * e1 + e2 * e2);
  const float inv = 1.f / angle;
  const float x = r0 * inv, y = r1 * inv, z = r2 * inv;
  const float s = sinf(angle), c = cosf(angle), oc = 1.f - c;
  float R[9];
  R[0] = 1.f + oc * (-z * z - y * y); R[1] = -s * z + oc * x * y; R[2] =  s * y + oc * x * z;
  R[3] =  s * z + oc * x * y;         R[4] = 1.f + oc * (-z * z - x * x); R[5] = -s * x + oc * y * z;
  R[6] = -s * y + oc * x * z;         R[7] =  s * x + oc * y * z; R[8] = 1.f + oc * (-y * y - x * x);
  float* ro = rot + (size_t)t * 9;
#pragma unroll
  for (int e = 0; e < 9; ++e) ro[e] = R[e];
  if (j >= 1) {
    float* p = F + (size_t)b * FW + 12 + (j - 1) * 9;
#pragma unroll
    for (int e = 0; e < 9; ++e)
      p[e] = R[e] - ((e == 0 || e == 4 || e == 8) ? 1.f : 0.f);
  }
}

// ---------------------------------------------------------------------------
// K2: per-batch kinematic chain -> rel transforms A (B,J,4,4) + posed joints
// ---------------------------------------------------------------------------
__global__ void k_chain(const float* __restrict__ betas,
                        const float* __restrict__ rot,
                        const float* __restrict__ Jt,
                        const float* __restrict__ JS,
                        float* __restrict__ A, float* __restrict__ outJ) {
  const int b = blockIdx.x * blockDim.x + threadIdx.x;
  if (b >= Bx) return;
  float bet[NBx];
#pragma unroll
  for (int l = 0; l < NBx; ++l) bet[l] = betas[b * NBx + l];
  float jr[Jx][3];
  for (int j = 0; j < Jx; ++j)
    for (int k = 0; k < 3; ++k) {
      float a = Jt[j * 3 + k];
      const float* js = JS + (j * 3 + k) * 10;
#pragma unroll
      for (int l = 0; l < NBx; ++l) a += js[l] * bet[l];
      jr[j][k] = a;
    }
  float* Ab = A + (size_t)b * Jx * 16;
  for (int i = 0; i < Jx; ++i) {
    const float* R = rot + ((size_t)b * Jx + i) * 9;
    const int p = d_parents[i];
    float rel[3];
    for (int k = 0; k < 3; ++k) rel[k] = (i == 0) ? jr[i][k] : jr[i][k] - jr[p][k];
    float* Ci = Ab + i * 16;
    if (i == 0) {
      for (int m = 0; m < 3; ++m) {
        Ci[m * 4 + 0] = R[m * 3 + 0]; Ci[m * 4 + 1] = R[m * 3 + 1];
        Ci[m * 4 + 2] = R[m * 3 + 2]; Ci[m * 4 + 3] = rel[m];
      }
    } else {
      const float* Cp = Ab + p * 16;
      for (int m = 0; m < 3; ++m) {
        const float p0 = Cp[m * 4 + 0], p1 = Cp[m * 4 + 1], p2 = Cp[m * 4 + 2], p3 = Cp[m * 4 + 3];
        Ci[m * 4 + 0] = p0 * R[0] + p1 * R[3] + p2 * R[6];
        Ci[m * 4 + 1] = p0 * R[1] + p1 * R[4] + p2 * R[7];
        Ci[m * 4 + 2] = p0 * R[2] + p1 * R[5] + p2 * R[8];
        Ci[m * 4 + 3] = p0 * rel[0] + p1 * rel[1] + p2 * rel[2] + p3;
      }
    }
    Ci[12] = 0.f; Ci[13] = 0.f; Ci[14] = 0.f; Ci[15] = 1.f;
    for (int k = 0; k < 3; ++k)
      outJ[((size_t)b * NJOUT + i) * 3 + k] = Ci[k * 4 + 3];   // posed joint
  }
  for (int i = 0; i < Jx; ++i) {
    float* Ci = Ab + i * 16;
    for (int m = 0; m < 3; ++m)
      Ci[m * 4 + 3] -= Ci[m * 4 + 0] * jr[i][0] + Ci[m * 4 + 1] * jr[i][1] +
                       Ci[m * 4 + 2] * jr[i][2];
  }
}

// ---------------------------------------------------------------------------
// K3: fused WMMA kernel, one wave per (16-batch x 16-vertex) tile.
// Hot loop is branch-free; only the 2 boundary chunks use clamp+mask selects.
// ---------------------------------------------------------------------------
#if HAVE_WMMA_F32X4
#define WMMA4(a, bf, c) \
  __builtin_amdgcn_wmma_f32_16x16x4_f32(false, (a), false, (bf), (short)0, (c), false, false)
#endif

__launch_bounds__(32)
__global__ void k_lbs(const float* __restrict__ F,
                      const float* __restrict__ sd,
                      const float* __restrict__ pd,
                      const float* __restrict__ vt,
                      const float* __restrict__ W,
                      const float* __restrict__ A,
                      float* __restrict__ verts) {
  __shared__ float sA[16 * 288];     // 16 batches x 24 joints x 12 (rows 0..2)
  __shared__ float sW[16 * 24];      // 16 vertices x 24 joints
  __shared__ float sP[3][16][17];    // v_posed tile: [coord][batch][vertex]

  const int lane = threadIdx.x;      // wave32
  const int half = lane >> 4;        // 0 | 1
  const int nIdx = lane & 15;
  const int v0 = blockIdx.x * 16;
  const int b0 = blockIdx.y * 16;
  const int vC = min(v0 + nIdx, Vx - 1);
  const int Vx3 = Vx * 3;

  // cooperative staging of A-tile and W-tile (overlaps WMMA issue)
  for (int i = lane; i < 16 * 288; i += 32) {
    const int bb = i / 288, r = i % 288, j = r / 12, e = r % 12;
    sA[i] = A[(((size_t)(b0 + bb)) * Jx + j) * 16 + e];
  }
  for (int i = lane; i < 16 * 24; i += 32) {
    const int vl = i / 24, j = i % 24;
    sW[i] = W[(size_t)min(v0 + vl, Vx - 1) * Jx + j];
  }

  const float bias0 = vt[vC * 3 + 0];
  const float bias1 = vt[vC * 3 + 1];
  const float bias2 = vt[vC * 3 + 2];

#if HAVE_WMMA_F32X4
  v8f c0 = {}, c1 = {}, c2 = {};
  const float* Frow   = F + (size_t)(b0 + nIdx) * FW;   // A-frag row M = nIdx
  const float* sdBase = sd + (size_t)vC * 30;           // B-frag col N = nIdx

  // ---- shape chunks 0..1 (l = 0..7): unconditional, b64-aligned loads ----
#pragma unroll
  for (int cc = 0; cc < 2; ++cc) {
    const int ka = cc * 4 + 2 * half;                   // even
    const v2f a  = *(const v2f*)(Frow + ka);
    const v2f bf0 = *(const v2f*)(sdBase + ka);
    const v2f bf1 = *(const v2f*)(sdBase + 10 + ka);
    const v2f bf2 = *(const v2f*)(sdBase + 20 + ka);
    c0 = WMMA4(a, bf0, c0);
    c1 = WMMA4(a, bf1, c1);
    c2 = WMMA4(a, bf2, c2);
  }
  { // ---- shape chunk 2 (l = 8..11): clamp + 0/1 mask, no branches ----
    const int ka = 8 + 2 * half;
    const v2f a = *(const v2f*)(Frow + ka);             // pads are zero
    const float m0 = (ka     < NBx) ? 1.f : 0.f;
    const float m1 = (ka + 1 < NBx) ? 1.f : 0.f;
    const int l0 = min(ka, NBx - 1), l1 = min(ka + 1, NBx - 1);
    v2f bf0, bf1, bf2;
    bf0.x = sdBase[l0]      * m0; bf0.y = sdBase[l1]      * m1;
    bf1.x = sdBase[10 + l0] * m0; bf1.y = sdBase[10 + l1] * m1;
    bf2.x = sdBase[20 + l0] * m0; bf2.y = sdBase[20 + l1] * m1;
    c0 = WMMA4(a, bf0, c0);
    c1 = WMMA4(a, bf1, c1);
    c2 = WMMA4(a, bf2, c2);
  }
  // ---- pose chunks 3..53 (p = 0..203): fully branch-free hot loop ----
  {
    const float* aP = Frow + 12 + 2 * half;             // even offset -> b64
    const float* r0 = pd + (size_t)(2 * half) * Vx3 + (size_t)vC * 3;
    for (int it = 0; it < 51; ++it) {
      const v2f a = *(const v2f*)aP;
      const float* r1 = r0 + Vx3;
      // prefetch ~2 chunks (8 rows) ahead; clamp address branchlessly
      const float* pfa = (it < 43) ? (r0 + 8 * (size_t)Vx3) : r0;
      __builtin_prefetch((const void*)pfa, 0, 1);
      v2f bf0, bf1, bf2;
      bf0.x = r0[0]; bf0.y = r1[0];
      bf1.x = r0[1]; bf1.y = r1[1];
      bf2.x = r0[2]; bf2.y = r1[2];
      c0 = WMMA4(a, bf0, c0);
      c1 = WMMA4(a, bf1, c1);
      c2 = WMMA4(a, bf2, c2);
      aP += 4;
      r0 += 4 * (size_t)Vx3;
    }
  }
  { // ---- pose chunk 54 (p = 204..207): clamp + mask the single pad row ----
    const int p = 204 + 2 * half;
    const v2f a = *(const v2f*)(Frow + 12 + p);         // F[219] is zero pad
    const float m1 = (p + 1 < Px) ? 1.f : 0.f;
    const int p1 = min(p + 1, Px - 1);
    const float* q0 = pd + (size_t)p  * Vx3 + (size_t)vC * 3;
    const float* q1 = pd + (size_t)p1 * Vx3 + (size_t)vC * 3;
    v2f bf0, bf1, bf2;
    bf0.x = q0[0]; bf0.y = q1[0] * m1;
    bf1.x = q0[1]; bf1.y = q1[1] * m1;
    bf2.x = q0[2]; bf2.y = q1[2] * m1;
    c0 = WMMA4(a, bf0, c0);
    c1 = WMMA4(a, bf1, c1);
    c2 = WMMA4(a, bf2, c2);
  }
  // spill C (M = r + 8*half, N = nIdx) to LDS, add v_template bias
#pragma unroll
  for (int r = 0; r < 8; ++r) {
    const int m = r + 8 * half;
    sP[0][m][nIdx] = c0[r] + bias0;
    sP[1][m][nIdx] = c1[r] + bias1;
    sP[2][m][nIdx] = c2[r] + bias2;
  }
#else
  // scalar fallback (also taken by the host compile pass)
  for (int r = 0; r < 8; ++r) {
    const int m = r + 8 * half;
    const float* Fr = F + (size_t)(b0 + m) * FW;
    float a0 = bias0, a1 = bias1, a2 = bias2;
    for (int l = 0; l < NBx; ++l) {
      const float bv = Fr[l];
      a0 += bv * sd[(size_t)vC * 30 + l];
      a1 += bv * sd[(size_t)vC * 30 + 10 + l];
      a2 += bv * sd[(size_t)vC * 30 + 20 + l];
    }
    for (int p = 0; p < Px; ++p) {
      const float pv = Fr[12 + p];
      const float* rr = pd + (size_t)p * (Vx * 3) + (size_t)vC * 3;
      a0 += pv * rr[0]; a1 += pv * rr[1]; a2 += pv * rr[2];
    }
    sP[0][m][nIdx] = a0; sP[1][m][nIdx] = a1; sP[2][m][nIdx] = a2;
  }
#endif
  __syncthreads();

  // ---- stage 2: LBS combine. Each lane: fixed vertex column, 8 batches ----
  float w[Jx];
#pragma unroll
  for (int j = 0; j < Jx; ++j) w[j] = sW[nIdx * 24 + j];
  const bool vok = (v0 + nIdx) < Vx;
#pragma unroll
  for (int i = 0; i < 8; ++i) {
    const int bloc = half + 2 * i;
    const float x = sP[0][bloc][nIdx];
    const float y = sP[1][bloc][nIdx];
    const float z = sP[2][bloc][nIdx];
    float a0 = 0.f, a1 = 0.f, a2 = 0.f;
#pragma unroll
    for (int j = 0; j < Jx; ++j) {
      const float* Ar = &sA[bloc * 288 + j * 12];   // 16B-aligned rows
      const float wj = w[j];
      a0 += wj * (Ar[0] * x + Ar[1] * y + Ar[2]  * z + Ar[3]);
      a1 += wj * (Ar[4] * x + Ar[5] * y + Ar[6]  * z + Ar[7]);
      a2 += wj * (Ar[8] * x + Ar[9] * y + Ar[10] * z + Ar[11]);
    }
    if (vok) {
      const size_t o = ((size_t)(b0 + bloc) * Vx + (v0 + nIdx)) * 3;
      verts[o + 0] = a0; verts[o + 1] = a1; verts[o + 2] = a2;
    }
  }
}

// ---------------------------------------------------------------------------
// K4: gather 11 extra joints
// ---------------------------------------------------------------------------
__global__ void k_extra11(const int* __restrict__ idxs,
                          const float* __restrict__ verts,
                          float* __restrict__ outJ) {
  const int t = blockIdx.x * blockDim.x + threadIdx.x;
  if (t >= Bx * 11 * 3) return;
  const int k = t % 3, r = t / 3, e = r % 11, b = r / 11;
  const int v = idxs[e];
  outJ[((size_t)b * NJOUT + 24 + e) * 3 + k] =
      verts[((size_t)b * Vx + v) * 3 + k];
}

// ---------------------------------------------------------------------------
// K5: extra9 + h36m17 joint regressions (verts + regressors are L2-resident)
// ---------------------------------------------------------------------------
__global__ void k_jreg(const float* __restrict__ R9,
                       const float* __restrict__ R17,
                       const float* __restrict__ verts,
                       float* __restrict__ outJ) {
  __shared__ float red[3 * 256];
  const int j = blockIdx.x;          // 0..25
  const int b = blockIdx.y;
  const float* row = (j < 9) ? (R9 + (size_t)j * Vx) : (R17 + (size_t)(j - 9) * Vx);
  const int jj = (j < 9) ? (35 + j) : (44 + (j - 9));
  const float* vb = verts + (size_t)b * Vx * 3;
  float a0 = 0.f, a1 = 0.f, a2 = 0.f;
  for (int v = threadIdx.x; v < Vx; v += 256) {
    const float wv = row[v];
    a0 += wv * vb[v * 3 + 0];
    a1 += wv * vb[v * 3 + 1];
    a2 += wv * vb[v * 3 + 2];
  }
  red[threadIdx.x] = a0; red[256 + threadIdx.x] = a1; red[512 + threadIdx.x] = a2;
  __syncthreads();
  for (int s = 128; s > 0; s >>= 1) {
    if (threadIdx.x < s) {
      red[threadIdx.x]       += red[threadIdx.x + s];
      red[256 + threadIdx.x] += red[256 + threadIdx.x + s];
      red[512 + threadIdx.x] += red[512 + threadIdx.x + s];
    }
    __syncthreads();
  }
  if (threadIdx.x == 0) {
    const size_t o = ((size_t)b * NJOUT + jj) * 3;
    outJ[o] = red[0]; outJ[o + 1] = red[256]; outJ[o + 2] = red[512];
  }
}

// ---------------------------------------------------------------------------
extern "C" void kernel_launch(void* const* d_in, const int* in_sizes, int n_in,
                              void* d_out, int out_size, void* d_ws, size_t ws_size,
                              hipStream_t stream) {
  const float* betas       = (const float*)d_in[0];
  const float* poses       = (const float*)d_in[1];
  const int*   extra_idxs  = (const int*)  d_in[2];
  const float* v_template  = (const float*)d_in[3];
  const float* shapedirs   = (const float*)d_in[4];
  const float* posedirs    = (const float*)d_in[5];
  const float* J_regressor = (const float*)d_in[6];
  const float* lbs_weights = (const float*)d_in[7];
  const float* Je9         = (const float*)d_in[8];
  const float* Jh17        = (const float*)d_in[9];

  float* out   = (float*)d_out;
  float* verts = out;                               // B*V*3
  float* outJ  = out + (size_t)Bx * Vx * 3;         // B*61*3

  // workspace layout (floats): ~3.36 MB total
  float* ws  = (float*)d_ws;
  float* rot = ws;                                  // B*J*9   = 221184
  float* F   = rot + (size_t)Bx * Jx * 9;           // B*220   = 225280
  float* A   = F   + (size_t)Bx * FW;               // B*J*16  = 393216
  float* Jt  = A   + (size_t)Bx * Jx * 16;          // 72
  float* JS  = Jt  + 72;                            // 720

  k_jcoef    <<<792, 256, 0, stream>>>(J_regressor, v_template, shapedirs, Jt, JS);
  k_feat     <<<(Bx * FW + 255) / 256, 256, 0, stream>>>(betas, F);
  k_rodrigues<<<(Bx * Jx + 255) / 256, 256, 0, stream>>>(poses, rot, F);
  k_chain    <<<(Bx + 255) / 256, 256, 0, stream>>>(betas, rot, Jt, JS, A, outJ);

  dim3 g4((Vx + 15) / 16, Bx / 16);                 // 431 x 64 tiles, 1 wave each
  k_lbs<<<g4, 32, 0, stream>>>(F, shapedirs, posedirs, v_template,
                               lbs_weights, A, verts);

  k_extra11<<<(Bx * 11 * 3 + 255) / 256, 256, 0, stream>>>(extra_idxs, verts, outJ);
  dim3 g5(26, Bx);
  k_jreg<<<g5, 256, 0, stream>>>(Je9, Jh17, verts, outJ);
}